// SphereNetLayer_37220186587493
// MI455X (gfx1250) — compile-verified
//
#include <hip/hip_runtime.h>
#include <hip/hip_bf16.h>

// ---------------------------------------------------------------------------
// SphereNet-style layer for MI455X (gfx1250, wave32, WMMA).
//
//   msg  = concat(rbf,ang) @ W_edge + b_edge          [E=160000, D=512], K=80
//   agg  = segment_sum(msg, row)                      [N=10000, D=512]
//   h    = relu(agg @ W1 + b1) @ W2 + b2
//   out  = x + h
//
// Strategy (roofline): fuse edge GEMM with the scatter using
// global_atomic_add_f32 so the 327MB msg tensor is never materialized; agg
// (20.5MB) lives in the 192MB L2 so atomics are L2-local. All three GEMMs run
// on V_WMMA_F32_16X16X4_F32 (fp32 tensor path, 16x16 C tiles, K=4/step).
// ---------------------------------------------------------------------------

typedef float v2f __attribute__((ext_vector_type(2)));
typedef float v8f __attribute__((ext_vector_type(8)));

#define WMMA_F32X4(a, b, c) \
    __builtin_amdgcn_wmma_f32_16x16x4_f32(false, (a), false, (b), (short)0, (c), false, false)

#define N_NODES 10000
#define N_EDGES 160000
#define EMBED   512
#define KEDGE   80   /* RBF(64) + ANG(16) */

// ---------------------------------------------------------------------------
// 0) zero the aggregation buffer
// ---------------------------------------------------------------------------
__global__ void __launch_bounds__(256) zero_kernel(float* __restrict__ p, int count) {
    int i = blockIdx.x * 256 + threadIdx.x;
    if (i < count) p[i] = 0.0f;
}

// ---------------------------------------------------------------------------
// 1) Edge GEMM + scatter-add.
//    Block = 128 threads (4 waves). blockIdx.x -> 16-column tile of EMBED,
//    blockIdx.y*4+wave -> 16-edge tile. W_edge[:,col:col+16] staged in LDS
//    (80x16 fp32 = 5KB) and reused by all 4 waves. K=80 -> 20 WMMA steps.
//    After accumulation each lane owns 8 C elements (rows m = v + 8*(lane>=16),
//    col n = lane&15); bias is folded in (segment_sum adds b_edge per edge)
//    and the element is atomically added to agg[row[e], col].
// ---------------------------------------------------------------------------
__global__ void __launch_bounds__(128) edge_scatter_kernel(
    const float* __restrict__ rbf,          // [E, 64]
    const float* __restrict__ ang,          // [E, 16]
    const long long* __restrict__ row,      // edge_index[0], int64 [E]
    const float* __restrict__ W_edge,       // [80, 512]
    const float* __restrict__ b_edge,       // [512]
    float* __restrict__ agg)                // [N, 512]
{
    __shared__ float wlds[KEDGE][16];

    const int tid     = threadIdx.x;
    const int colBase = blockIdx.x * 16;

    // stage W_edge column tile: 80x16 floats, 128 threads -> 10 each
    for (int i = tid; i < KEDGE * 16; i += 128) {
        int k = i >> 4, n = i & 15;
        wlds[k][n] = W_edge[k * EMBED + colBase + n];
    }
    __syncthreads();

    const int wid  = tid >> 5;
    const int lane = tid & 31;
    const int e0   = (blockIdx.y * 4 + wid) * 16;   // 16-edge tile base
    const int m    = lane & 15;                     // A: M index / B: N index
    const int kofs = (lane >> 4) << 1;              // lanes 16-31 hold K+2,K+3
    const int e    = e0 + m;

    v8f c = {};
    #pragma unroll
    for (int kk = 0; kk < KEDGE / 4; ++kk) {
        const int k = kk * 4 + kofs;                // even; pair never crosses 64
        v2f a;
        if (k < 64) {
            a = *(const v2f*)(rbf + (long)e * 64 + k);
        } else {
            a = *(const v2f*)(ang + (long)e * 16 + (k - 64));
        }
        v2f b;
        b.x = wlds[k][m];
        b.y = wlds[k + 1][m];
        c = WMMA_F32X4(a, b, c);
    }

    const int   n    = lane & 15;
    const float bias = b_edge[colBase + n];
    const int   mhi  = (lane >> 4) * 8;
    #pragma unroll
    for (int v = 0; v < 8; ++v) {
        const long long r = row[e0 + mhi + v];
        atomicAdd(&agg[r * EMBED + colBase + n], c[v] + bias);
    }
}

// ---------------------------------------------------------------------------
// 2) Dense GEMM  out[M,512] = act(A[M,512] @ W[512,512] + bias) (+ addend)
//    Block = 256 threads (8 waves); each block owns a 16-row strip, each wave
//    a 16x64 tile (4 v8f accumulators). A-tile (16x64) staged in LDS with +4
//    padding (stride 68 -> lanes 0..15 hit distinct banks). W read directly
//    from global: W1/W2 are 1MB each and stay L2-resident (192MB L2).
// ---------------------------------------------------------------------------
__global__ void __launch_bounds__(256) gemm_bias_act_kernel(
    const float* __restrict__ A,        // [M, 512]
    const float* __restrict__ W,        // [512, 512]
    const float* __restrict__ bias,     // [512]
    const float* __restrict__ addend,   // [M, 512] or nullptr
    float* __restrict__ out,            // [M, 512]
    int do_relu)
{
    __shared__ float aT[16][68];

    const int tid   = threadIdx.x;
    const int mBase = blockIdx.x * 16;
    const int wid   = tid >> 5;
    const int lane  = tid & 31;
    const int colW  = wid * 64;                 // wave's 64-column slab
    const int m     = lane & 15;
    const int kofs  = (lane >> 4) << 1;

    v8f cacc[4] = {};

    for (int kb = 0; kb < EMBED; kb += 64) {
        __syncthreads();
        for (int i = tid; i < 16 * 64; i += 256) {
            int r = i >> 6, kk = i & 63;
            aT[r][kk] = A[(long)(mBase + r) * EMBED + kb + kk];
        }
        __syncthreads();

        #pragma unroll
        for (int ks = 0; ks < 16; ++ks) {
            const int k0 = ks * 4 + kofs;
            v2f a;
            a.x = aT[m][k0];
            a.y = aT[m][k0 + 1];

            const float* wp = W + (long)(kb + k0) * EMBED + colW + m;
            #pragma unroll
            for (int f = 0; f < 4; ++f) {
                v2f b;
                b.x = wp[f * 16];
                b.y = wp[f * 16 + EMBED];
                cacc[f] = WMMA_F32X4(a, b, cacc[f]);
            }
        }
    }

    const int n   = lane & 15;
    const int mhi = (lane >> 4) * 8;
    #pragma unroll
    for (int f = 0; f < 4; ++f) {
        const int   col = colW + f * 16 + n;
        const float bv  = bias[col];
        #pragma unroll
        for (int v = 0; v < 8; ++v) {
            const long rowI = mBase + mhi + v;
            float val = cacc[f][v] + bv;
            if (do_relu) val = fmaxf(val, 0.0f);
            if (addend)  val += addend[rowI * EMBED + col];
            out[rowI * EMBED + col] = val;
        }
    }
}

// ---------------------------------------------------------------------------
// Launch: zero agg -> edge scatter -> gemm1(relu) -> gemm2(+x)
// Workspace: agg [10000*512] f32 @ 0, h1 [10000*512] f32 after it (~41MB).
// ---------------------------------------------------------------------------
extern "C" void kernel_launch(void* const* d_in, const int* in_sizes, int n_in,
                              void* d_out, int out_size, void* d_ws, size_t ws_size,
                              hipStream_t stream) {
    const float*     x          = (const float*)d_in[0];
    /* d_in[1] = coord, unused by the reference */
    const long long* edge_index = (const long long*)d_in[2];   // [2, E] int64
    const float*     rbf        = (const float*)d_in[3];
    const float*     ang        = (const float*)d_in[4];
    const float*     W_edge     = (const float*)d_in[5];
    const float*     b_edge     = (const float*)d_in[6];
    const float*     W1         = (const float*)d_in[7];
    const float*     b1         = (const float*)d_in[8];
    const float*     W2         = (const float*)d_in[9];
    const float*     b2         = (const float*)d_in[10];
    float*           out        = (float*)d_out;

    float* agg = (float*)d_ws;
    float* h1  = agg + (size_t)N_NODES * EMBED;

    const long long* row = edge_index;   // edge_index[0] = first E entries

    const int aggElems = N_NODES * EMBED;
    zero_kernel<<<(aggElems + 255) / 256, 256, 0, stream>>>(agg, aggElems);

    // 32 column tiles x (160000/16/4 = 2500) edge-tile blocks
    edge_scatter_kernel<<<dim3(EMBED / 16, N_EDGES / 16 / 4), 128, 0, stream>>>(
        rbf, ang, row, W_edge, b_edge, agg);

    // h1 = relu(agg @ W1 + b1)
    gemm_bias_act_kernel<<<N_NODES / 16, 256, 0, stream>>>(agg, W1, b1, nullptr, h1, 1);

    // out = x + (h1 @ W2 + b2)
    gemm_bias_act_kernel<<<N_NODES / 16, 256, 0, stream>>>(h1, W2, b2, x, out, 0);
}